// PCDRefinement_62362925138478
// MI455X (gfx1250) — compile-verified
//
#include <hip/hip_runtime.h>
#include <math.h>

typedef float v2f __attribute__((ext_vector_type(2)));
typedef float v8f __attribute__((ext_vector_type(8)));

#define BB   4
#define NN   4096
#define DD   67      // 3 + 64
#define DP   68      // padded K (17 * 4)
#define CC   64
#define KNN  16
#define NTIL (NN / 16)

// ---------------------------------------------------------------------------
// Build x = [loc;feat]^T rows (padded to DP) and squared norms.
// ---------------------------------------------------------------------------
__global__ __launch_bounds__(256) void k_buildx(const float* __restrict__ loc,
                                                const float* __restrict__ feat,
                                                float* __restrict__ xbuf,
                                                float* __restrict__ sq) {
    int t = blockIdx.x * blockDim.x + threadIdx.x;   // t = b*NN + n
    if (t >= BB * NN) return;
    int b = t / NN, n = t % NN;
    float s = 0.f;
    float* xr = xbuf + (size_t)t * DP;
#pragma unroll
    for (int d = 0; d < 3; ++d) {
        float v = loc[(b * 3 + d) * NN + n];
        xr[d] = v; s += v * v;
    }
    for (int c = 0; c < CC; ++c) {
        float v = feat[((size_t)b * CC + c) * NN + n];
        xr[3 + c] = v; s += v * v;
    }
    xr[DD] = 0.f;                 // zero pad so K=68 WMMA loop is exact
    sq[t] = s;
}

// ---------------------------------------------------------------------------
// kNN via WMMA f32 Gram tiles + per-row streaming top-16.
// One wave32 per 32-row tile (two stacked 16x16 WMMA accumulators sharing the
// B operand); blockIdx.x = 32-row tile, blockIdx.y = batch.
// ---------------------------------------------------------------------------
__global__ __launch_bounds__(32) void k_knn(const float* __restrict__ xbuf,
                                            const float* __restrict__ sq,
                                            int* __restrict__ idx) {
    __shared__ __align__(16) float asmem[32 * DP];
    __shared__ __align__(16) float bsmem[16 * DP];
    __shared__ float dts[32 * 16];
    __shared__ float sqi[32];

    const int lane = threadIdx.x;
    const int b    = blockIdx.y;
    const int i0   = blockIdx.x * 32;
    const float* xb  = xbuf + (size_t)b * NN * DP;
    const float* sqb = sq + (size_t)b * NN;

    // Stage A tile (32 rows x 68 floats) once, 128b at a time.
    {
        const float4* src = (const float4*)(xb + (size_t)i0 * DP);
        float4* dst = (float4*)asmem;
        for (int t = lane; t < 32 * (DP / 4); t += 32) dst[t] = src[t];
        sqi[lane] = sqb[i0 + lane];
    }
    __syncthreads();

    float bestd[KNN];
    int   besti[KNN];
#pragma unroll
    for (int t = 0; t < KNN; ++t) { bestd[t] = 3.0e38f; besti[t] = 0; }

    const int mrow = lane & 15;   // col within tile / row within 16-row half
    const int hi   = lane >> 4;   // lane half

    for (int jt = 0; jt < NTIL; ++jt) {
        const int j0 = jt * 16;

        // Stage B tile (16 rows x 68 floats), 128b at a time.
        {
            const float4* src = (const float4*)(xb + (size_t)j0 * DP);
            float4* dst = (float4*)bsmem;
            for (int t = lane; t < 16 * (DP / 4); t += 32) dst[t] = src[t];
        }
        // Prefetch next column tile (lane-strided 128B covers the tile's lines).
        if (jt + 1 < NTIL) {
            const float* nsrc = xb + (size_t)(j0 + 16) * DP;
            __builtin_prefetch(nsrc + lane * 32, 0, 3);
        }
        __syncthreads();

        // Two stacked Gram tiles: 2x17 V_WMMA_F32_16X16X4_F32 sharing B reads.
        v8f acc0 = {}, acc1 = {};
#pragma unroll
        for (int k0 = 0; k0 < DP; k0 += 4) {
            v2f a0, a1, bv;
            // B 4x16 layout (B = X^T): v0 -> K rows k0 (lo half) / k0+1 (hi),
            //                          v1 -> K rows k0+2 / k0+3, N = lane&15
            bv.x = bsmem[mrow * DP + k0 + hi];
            bv.y = bsmem[mrow * DP + k0 + 2 + hi];
            // A 16x4 layout: lanes 0-15 -> M=lane, K=k0/k0+1; lanes 16-31 -> K=k0+2/k0+3
            a0.x = asmem[mrow * DP + k0 + hi * 2];
            a0.y = asmem[mrow * DP + k0 + hi * 2 + 1];
            a1.x = asmem[(mrow + 16) * DP + k0 + hi * 2];
            a1.y = asmem[(mrow + 16) * DP + k0 + hi * 2 + 1];
            acc0 = __builtin_amdgcn_wmma_f32_16x16x4_f32(
                false, a0, false, bv, (short)0, acc0, false, false);
            acc1 = __builtin_amdgcn_wmma_f32_16x16x4_f32(
                false, a1, false, bv, (short)0, acc1, false, false);
        }

        // Distances into LDS: C/D layout -> VGPR v holds row v (lanes 0-15) / v+8.
        const float sqj = sqb[j0 + mrow];
#pragma unroll
        for (int v = 0; v < 8; ++v) {
            int r0 = v + hi * 8;
            float d0 = sqi[r0] - 2.0f * acc0[v] + sqj;
            if (i0 + r0 == j0 + mrow) d0 += 1.0e9f;   // exclude self-loop
            dts[r0 * 16 + mrow] = d0;
            int r1 = r0 + 16;
            float d1 = sqi[r1] - 2.0f * acc1[v] + sqj;
            if (i0 + r1 == j0 + mrow) d1 += 1.0e9f;
            dts[r1 * 16 + mrow] = d1;
        }
        __syncthreads();

        // Every lane owns one of the 32 rows; sorted insert (fully unrolled).
#pragma unroll
        for (int cc = 0; cc < 16; ++cc) {
            float c  = dts[lane * 16 + cc];
            int   ci = j0 + cc;
            if (c < bestd[KNN - 1]) {
#pragma unroll
                for (int t2 = KNN - 1; t2 >= 0; --t2) {
                    float pd = (t2 == 0) ? -3.0e38f : bestd[t2 - 1];
                    int   pi = (t2 == 0) ? 0        : besti[t2 - 1];
                    if (pd > c)             { bestd[t2] = pd; besti[t2] = pi; }
                    else if (bestd[t2] > c) { bestd[t2] = c;  besti[t2] = ci; }
                }
            }
        }
        __syncthreads();
    }

    {
        int* op = idx + ((size_t)b * NN + i0 + lane) * KNN;
#pragma unroll
        for (int t = 0; t < KNN; ++t) op[t] = besti[t];
    }
}

// ---------------------------------------------------------------------------
// GraphConv: out[c] = relu(agg . w_rel[c] + x . w_root[c] + bias[c]).
// One block (64 threads) per point; agg staged in LDS.
// ---------------------------------------------------------------------------
__global__ __launch_bounds__(64) void k_conv(const float* __restrict__ xbuf,
                                             const int* __restrict__ idx,
                                             const float* __restrict__ wrel,
                                             const float* __restrict__ wroot,
                                             const float* __restrict__ bias,
                                             float* __restrict__ fout) {
    __shared__ float agg[DP];
    __shared__ float xrow[DP];
    __shared__ int   idxs[KNN];
    const int t  = threadIdx.x;
    const int bn = blockIdx.x;                // b*NN + n
    const int b  = bn / NN, n = bn % NN;
    const float* xb = xbuf + (size_t)b * NN * DP;

    if (t < KNN) idxs[t] = idx[(size_t)bn * KNN + t];
    __syncthreads();

    for (int d = t; d < DD; d += 64) {
        float s = 0.f;
#pragma unroll
        for (int k = 0; k < KNN; ++k) s += xb[(size_t)idxs[k] * DP + d];
        agg[d]  = s;
        xrow[d] = xbuf[(size_t)bn * DP + d];
    }
    __syncthreads();

    float acc = bias[t];
    for (int d = 0; d < DD; ++d)
        acc = fmaf(agg[d], wrel[t * DD + d], fmaf(xrow[d], wroot[t * DD + d], acc));
    fout[((size_t)b * CC + t) * NN + n] = fmaxf(acc, 0.f);
}

// ---------------------------------------------------------------------------
// Final 1x1 conv + tanh + residual; also copy feat into output slot 2.
// ---------------------------------------------------------------------------
__global__ __launch_bounds__(256) void k_final(const float* __restrict__ loc,
                                               const float* __restrict__ feat,
                                               const float* __restrict__ locw,
                                               const float* __restrict__ locb,
                                               float* __restrict__ out) {
    int t = blockIdx.x * blockDim.x + threadIdx.x;
    if (t >= BB * NN) return;
    int b = t / NN, n = t % NN;
    float l[3], s[3];
#pragma unroll
    for (int d = 0; d < 3; ++d) l[d] = loc[(b * 3 + d) * NN + n];
#pragma unroll
    for (int o = 0; o < 3; ++o)
        s[o] = locb[o] + locw[o * DD + 0] * l[0] + locw[o * DD + 1] * l[1]
                       + locw[o * DD + 2] * l[2];
    const size_t featBase = (size_t)BB * 3 * NN;
    for (int c = 0; c < CC; ++c) {
        float f = feat[((size_t)b * CC + c) * NN + n];
#pragma unroll
        for (int o = 0; o < 3; ++o) s[o] += locw[o * DD + 3 + c] * f;
        out[featBase + ((size_t)b * CC + c) * NN + n] = f;
    }
#pragma unroll
    for (int o = 0; o < 3; ++o)
        out[((size_t)b * 3 + o) * NN + n] = l[o] + tanhf(s[o]);
}

// ---------------------------------------------------------------------------
extern "C" void kernel_launch(void* const* d_in, const int* in_sizes, int n_in,
                              void* d_out, int out_size, void* d_ws, size_t ws_size,
                              hipStream_t stream) {
    const float* loc    = (const float*)d_in[0];
    const float* feat0  = (const float*)d_in[1];
    const float* w_rel0 = (const float*)d_in[2];
    const float* w_rt0  = (const float*)d_in[3];
    const float* b0     = (const float*)d_in[4];
    const float* w_relm = (const float*)d_in[5];
    const float* w_rtm  = (const float*)d_in[6];
    const float* bm     = (const float*)d_in[7];
    const float* w_rell = (const float*)d_in[8];
    const float* w_rtl  = (const float*)d_in[9];
    const float* bl     = (const float*)d_in[10];
    const float* loc_w  = (const float*)d_in[11];
    const float* loc_b  = (const float*)d_in[12];

    char* ws = (char*)d_ws;
    size_t off = 0;
    float* xbuf = (float*)(ws + off); off += (size_t)BB * NN * DP * sizeof(float);
    float* sqb  = (float*)(ws + off); off += (size_t)BB * NN * sizeof(float);
    int*   idxb = (int*)  (ws + off); off += (size_t)BB * NN * KNN * sizeof(int);
    float* fA   = (float*)(ws + off); off += (size_t)BB * CC * NN * sizeof(float);
    float* fB   = (float*)(ws + off); off += (size_t)BB * CC * NN * sizeof(float);
    (void)ws_size; (void)in_sizes; (void)n_in; (void)out_size;

    const int ptBlocks = (BB * NN + 255) / 256;
    dim3 knnGrid(NN / 32, BB);

    auto layer = [&](const float* featIn, const float* wr, const float* wo,
                     const float* bi, float* featOut) {
        k_buildx<<<ptBlocks, 256, 0, stream>>>(loc, featIn, xbuf, sqb);
        k_knn   <<<knnGrid, 32, 0, stream>>>(xbuf, sqb, idxb);
        k_conv  <<<BB * NN, 64, 0, stream>>>(xbuf, idxb, wr, wo, bi, featOut);
    };

    layer(feat0, w_rel0, w_rt0, b0, fA);   // first layer
    layer(fA,    w_relm, w_rtm, bm, fB);   // mid layer 1 (shared weights)
    layer(fB,    w_relm, w_rtm, bm, fA);   // mid layer 2 (shared weights)
    layer(fA,    w_rell, w_rtl, bl, fB);   // last conv layer

    k_final<<<ptBlocks, 256, 0, stream>>>(loc, fB, loc_w, loc_b, (float*)d_out);
}